// AttBlock_8297876816055
// MI455X (gfx1250) — compile-verified
//
#include <hip/hip_runtime.h>
#include <hip/hip_bf16.h>

// Problem constants (from reference): BS=4, C=64, H=W=64, HW=4096, C8=8.
#define BS 4
#define C  64
#define HW 4096
#define C8 8
#define SROWS 80   // stacked projection rows: 64 (v) + 8 (k) + 8 (q_l)

typedef __attribute__((ext_vector_type(2))) float v2f;
typedef __attribute__((ext_vector_type(8))) float v8f;

// -----------------------------------------------------------------------------
// Kernel 1: S[b, 0:80, 0:4096] = [v_w; k_w; q_l_w] (80x64) @ x[b] (64x4096) + bias
// One wave computes one 16x16 output tile with 16x V_WMMA_F32_16X16X4_F32 (K=64).
// grid = (32 n-groups of 8 tiles, 5 m-tiles, 4 batches), 256 threads (8 waves).
// -----------------------------------------------------------------------------
__global__ __launch_bounds__(256) void k1_proj_wmma(
    const float* __restrict__ x,                       // [4,64,4096]
    const float* __restrict__ vw, const float* __restrict__ vb,
    const float* __restrict__ kw, const float* __restrict__ kb,
    const float* __restrict__ qw, const float* __restrict__ qb,
    float* __restrict__ S)                             // [4,80,4096]
{
    const int lane = threadIdx.x & 31;
    const int wave = threadIdx.x >> 5;
    const int n0   = (blockIdx.x * 8 + wave) * 16;     // spatial tile
    const int m0   = blockIdx.y * 16;                  // stacked-row tile (0..79)
    const int b    = blockIdx.z;

    const int half = lane >> 4;                        // 0: K pair {0,1}; 1: {2,3}
    const int l15  = lane & 15;
    const int arow = m0 + l15;

    const float* xb = x + (size_t)b * C * HW;

    // Preload all 16 A fragments (weights are tiny, stay in registers).
    v2f afrag[16];
#pragma unroll
    for (int ks = 0; ks < 16; ++ks) {
        const int kbase = ks * 4 + half * 2;
        float a0, a1;
        if (arow < 64)      { a0 = vw[arow * 64 + kbase];        a1 = vw[arow * 64 + kbase + 1]; }
        else if (arow < 72) { a0 = kw[(arow - 64) * 64 + kbase]; a1 = kw[(arow - 64) * 64 + kbase + 1]; }
        else                { a0 = qw[(arow - 72) * 64 + kbase]; a1 = qw[(arow - 72) * 64 + kbase + 1]; }
        afrag[ks].x = a0; afrag[ks].y = a1;
    }

    v8f acc = {};
#pragma unroll
    for (int ks = 0; ks < 16; ++ks) {
        const int kbase = ks * 4 + half * 2;
        v2f bfrag;
        bfrag.x = xb[(size_t)kbase * HW + n0 + l15];
        bfrag.y = xb[(size_t)(kbase + 1) * HW + n0 + l15];
        acc = __builtin_amdgcn_wmma_f32_16x16x4_f32(
            /*neg_a=*/false, afrag[ks], /*neg_b=*/false, bfrag,
            /*c_mod=*/(short)0, acc, /*reuse_a=*/false, /*reuse_b=*/false);
    }

    float* Sb = S + (size_t)b * SROWS * HW;
#pragma unroll
    for (int i = 0; i < 8; ++i) {
        const int M = m0 + i + half * 8;               // C/D layout: VGPR i -> M=i (+8 for hi half)
        const float bias = (M < 64) ? vb[M] : (M < 72 ? kb[M - 64] : qb[M - 72]);
        Sb[(size_t)M * HW + n0 + l15] = acc[i] + bias;
    }
}

// -----------------------------------------------------------------------------
// Kernel 2: rank-8 reductions over q = 0..4095.
//   M1[b,c,e] = sum_q v[b,c,q] * qg_w[q,e]
//   M2[b,c,e] = sum_q v[b,c,q] * k[b,e,q]
//   s [b,c]   = sum_q v[b,c,q] * qg_b[q]
// Folds gammas: G[b,c,0:8] = g_g*M1, G[b,c,8:16] = g_l*M2, B2[b,c] = g_g*s.
// grid = (64 channels, 4 batches), 256 threads, LDS tree reduction.
// -----------------------------------------------------------------------------
__global__ __launch_bounds__(256) void k2_reduce(
    const float* __restrict__ S,                       // [4,80,4096]
    const float* __restrict__ qgw,                     // [4096,8]
    const float* __restrict__ qgb,                     // [4096]
    const float* __restrict__ gamma_g,
    const float* __restrict__ gamma_l,
    float* __restrict__ G,                             // [4,64,16]
    float* __restrict__ B2)                            // [4,64]
{
    const int c   = blockIdx.x;
    const int b   = blockIdx.y;
    const int tid = threadIdx.x;

    const float* Sb   = S + (size_t)b * SROWS * HW;
    const float* vrow = Sb + (size_t)c * HW;

    float a1[8] = {}, a2[8] = {};
    float as = 0.f;
    for (int q = tid; q < HW; q += 256) {
        const float vq = vrow[q];
#pragma unroll
        for (int e = 0; e < 8; ++e) a1[e] += vq * qgw[q * 8 + e];
#pragma unroll
        for (int e = 0; e < 8; ++e) a2[e] += vq * Sb[(size_t)(64 + e) * HW + q];
        as += vq * qgb[q];
    }

    __shared__ float red[256];
    const float gg = gamma_g[0];
    const float gl = gamma_l[0];
    float* Gout = G + ((size_t)b * 64 + c) * 16;

    for (int i = 0; i < 17; ++i) {
        const float val = (i < 8) ? a1[i] : (i < 16 ? a2[i - 8] : as);
        red[tid] = val;
        __syncthreads();
        for (int off = 128; off > 0; off >>= 1) {
            if (tid < off) red[tid] += red[tid + off];
            __syncthreads();
        }
        if (tid == 0) {
            const float r = red[0];
            if (i < 8)        Gout[i] = gg * r;
            else if (i < 16)  Gout[i] = gl * r;
            else              B2[b * 64 + c] = gg * r;
        }
        __syncthreads();
    }
}

// -----------------------------------------------------------------------------
// Kernel 3: out[b,c,p] = sum_{e<16} G[b,c,e]*Bv[b,e,p] + B2[b,c] + x[b,c,p]
//   Bv[e<8 , p] = k_flat[b][p*8 + e]     (raw .view() regroup of k)
//   Bv[e>=8, p] = q_l_pre[b][e-8][p]     (transpose of q_l conv output)
// One wave per 16x16 tile, K=16 -> 4x V_WMMA_F32_16X16X4_F32.
// grid = (32 n-groups, 4 m-tiles, 4 batches), 256 threads.
// -----------------------------------------------------------------------------
__global__ __launch_bounds__(256) void k3_expand_wmma(
    const float* __restrict__ S,                       // [4,80,4096]
    const float* __restrict__ G,                       // [4,64,16]
    const float* __restrict__ B2,                      // [4,64]
    const float* __restrict__ x,                       // [4,64,4096]
    float* __restrict__ out)                           // [4,64,4096]
{
    const int lane = threadIdx.x & 31;
    const int wave = threadIdx.x >> 5;
    const int n0   = (blockIdx.x * 8 + wave) * 16;
    const int m0   = blockIdx.y * 16;                  // 0..63
    const int b    = blockIdx.z;
    const int half = lane >> 4;
    const int l15  = lane & 15;

    const float* Gb    = G + (size_t)b * 64 * 16;
    const float* Sb    = S + (size_t)b * SROWS * HW;
    const float* kflat = Sb + (size_t)64 * HW;         // k rows, 8*4096 contiguous
    const float* qlp   = Sb + (size_t)72 * HW;         // q_l_pre [8,4096]

    v2f afrag[4];
#pragma unroll
    for (int ks = 0; ks < 4; ++ks) {
        const int kbase = ks * 4 + half * 2;
        afrag[ks].x = Gb[(m0 + l15) * 16 + kbase];
        afrag[ks].y = Gb[(m0 + l15) * 16 + kbase + 1];
    }

    const int p = n0 + l15;
    v8f acc = {};
#pragma unroll
    for (int ks = 0; ks < 4; ++ks) {
        const int k0 = ks * 4 + half * 2;
        const int k1 = k0 + 1;
        v2f bfrag;
        bfrag.x = (k0 < 8) ? kflat[(size_t)p * 8 + k0] : qlp[(size_t)(k0 - 8) * HW + p];
        bfrag.y = (k1 < 8) ? kflat[(size_t)p * 8 + k1] : qlp[(size_t)(k1 - 8) * HW + p];
        acc = __builtin_amdgcn_wmma_f32_16x16x4_f32(
            false, afrag[ks], false, bfrag, (short)0, acc, false, false);
    }

    const float* xb = x + (size_t)b * C * HW;
    float* ob       = out + (size_t)b * C * HW;
#pragma unroll
    for (int i = 0; i < 8; ++i) {
        const int M = m0 + i + half * 8;
        const size_t idx = (size_t)M * HW + p;
        ob[idx] = acc[i] + B2[b * 64 + M] + xb[idx];
    }
}

// -----------------------------------------------------------------------------
// Launcher. Input order: x, q_l_w, q_l_b, k_w, k_b, v_w, v_b, qg_w, qg_b,
//                        gamma_g, gamma_l
// -----------------------------------------------------------------------------
extern "C" void kernel_launch(void* const* d_in, const int* in_sizes, int n_in,
                              void* d_out, int out_size, void* d_ws, size_t ws_size,
                              hipStream_t stream) {
    const float* x       = (const float*)d_in[0];
    const float* q_l_w   = (const float*)d_in[1];
    const float* q_l_b   = (const float*)d_in[2];
    const float* k_w     = (const float*)d_in[3];
    const float* k_b     = (const float*)d_in[4];
    const float* v_w     = (const float*)d_in[5];
    const float* v_b     = (const float*)d_in[6];
    const float* qg_w    = (const float*)d_in[7];
    const float* qg_b    = (const float*)d_in[8];
    const float* gamma_g = (const float*)d_in[9];
    const float* gamma_l = (const float*)d_in[10];
    float* out = (float*)d_out;

    // Workspace layout (floats): S [4*80*4096] | G [4*64*16] | B2 [4*64]
    float* S  = (float*)d_ws;
    float* G  = S + (size_t)BS * SROWS * HW;
    float* B2 = G + (size_t)BS * 64 * 16;

    // 1) stacked 1x1-conv projections: S = [v; k; q_l]
    k1_proj_wmma<<<dim3(HW / (8 * 16), SROWS / 16, BS), 256, 0, stream>>>(
        x, v_w, v_b, k_w, k_b, q_l_w, q_l_b, S);

    // 2) rank-8 contractions (M1, M2, s) with gammas folded in
    k2_reduce<<<dim3(64, BS), 256, 0, stream>>>(
        S, qg_w, qg_b, gamma_g, gamma_l, G, B2);

    // 3) rank-16 expansion + bias + residual
    k3_expand_wmma<<<dim3(HW / (8 * 16), 64 / 16, BS), 256, 0, stream>>>(
        S, G, B2, x, out);
}